// EdgeConv_61435212202233
// MI455X (gfx1250) — compile-verified
//
#include <hip/hip_runtime.h>
#include <hip/hip_bf16.h>
#include <math.h>

// ---------------------------------------------------------------------------
// EdgeConv, restructured:
//   h[b,n,e,:] = (x_n @ W1 - x_n @ W2 + b) + x_{idx} @ W2
//   out[b,n,:] = ELU( Z1[n] - Z2[n] + b + max_{valid e} Z2[idx_e] )
// Z1/Z2 computed with V_WMMA_F32_16X16X4_F32 (full fp32 precision).
// W is pre-packed into per-lane WMMA fragment order so the GEMM B operand is
// a single coalesced global_load_b64 per WMMA.
// ---------------------------------------------------------------------------

typedef __attribute__((ext_vector_type(2))) float v2f;
typedef __attribute__((ext_vector_type(8))) float v8f;

#define B_DIM 4
#define N_DIM 10000
#define C_DIM 128
#define O_DIM 128
#define E_DIM 32
#define NOUT  256          // Z row = [Z1 (128) | Z2 (128)]
#define M_DIM (B_DIM * N_DIM)
#define NTILES 16          // 256 output cols / 16
#define KSTEPS 32          // 128 / 4

// ---------------------------------------------------------------------------
// Pack W[256,128] into WMMA B-fragment order:
//   Bp[((nt*KSTEPS + ks)*32 + lane)] = { W[kk+rowoff, o], W[kk+1+rowoff, o] }
//   kk = 4*ks + 2*(lane>>4),  n = nt*16 + (lane&15),
//   rowoff = (n>=128)?128:0 (W2 block), o = n & 127.
// 16*32*32 = 16384 v2f entries (128 KB).
// ---------------------------------------------------------------------------
__global__ __launch_bounds__(256)
void edgeconv_packB(const float* __restrict__ W, v2f* __restrict__ Bp) {
  const int i = blockIdx.x * 256 + threadIdx.x;
  if (i >= NTILES * KSTEPS * 32) return;
  const int lane = i & 31;
  const int ks   = (i >> 5) & (KSTEPS - 1);
  const int nt   = i >> 10;
  const int half = lane >> 4;
  const int l16  = lane & 15;
  const int n    = nt * 16 + l16;
  const int rowoff = (n >= O_DIM) ? C_DIM : 0;
  const int o      = n & (O_DIM - 1);
  const int kk     = 4 * ks + 2 * half;
  v2f v;
  v.x = W[(size_t)(kk + rowoff)     * O_DIM + o];
  v.y = W[(size_t)(kk + 1 + rowoff) * O_DIM + o];
  Bp[i] = v;
}

// ---------------------------------------------------------------------------
// GEMM: Z[m, 0:128]   = x[m,:] @ W[0:128,:]
//       Z[m, 128:256] = x[m,:] @ W[128:256,:]
// block = 256 threads (8 waves); block owns 16 rows x 256 cols.
// wave w owns cols [w*32, w*32+32) as two 16x16 WMMA tiles, K-loop step 4.
// A from LDS (ds_load b64 pairs), B from packed Bp (one b64 per WMMA).
// ---------------------------------------------------------------------------
__global__ __launch_bounds__(256)
void edgeconv_gemm(const float* __restrict__ x, const v2f* __restrict__ Bp,
                   float* __restrict__ Z) {
  __shared__ float ldsA[16 * C_DIM];   // 8 KB A tile

  const int tid = threadIdx.x;
  const int m0  = blockIdx.x * 16;

  // Stage A tile (16x128 fp32 = 512 float4) coalesced: 256 thr x 2 rounds.
  const float4* xsrc = (const float4*)(x + (size_t)m0 * C_DIM);
  float4* adst = (float4*)ldsA;
  adst[tid]       = xsrc[tid];
  adst[tid + 256] = xsrc[tid + 256];
  __syncthreads();

  const int wave = tid >> 5;
  const int lane = tid & 31;
  const int half = lane >> 4;    // 0: lanes 0-15, 1: lanes 16-31
  const int l16  = lane & 15;

  #pragma unroll
  for (int tt = 0; tt < 2; ++tt) {
    const int nt = wave * 2 + tt;                     // n-tile index 0..15
    const v2f* Bt = Bp + (size_t)nt * KSTEPS * 32 + lane;

    v8f acc = {};
    for (int ks = 0; ks < KSTEPS; ++ks) {
      const int kk = 4 * ks + 2 * half;               // A: V0=K{0,2}, V1=K{1,3}
      v2f a;
      a.x = ldsA[l16 * C_DIM + kk];
      a.y = ldsA[l16 * C_DIM + kk + 1];
      const v2f bf = Bt[ks * 32];                     // coalesced b64, frag order
      acc = __builtin_amdgcn_wmma_f32_16x16x4_f32(
                false, a, false, bf, (short)0, acc, false, false);
    }

    // C/D layout: lanes 0-15 -> rows r, lanes 16-31 -> rows r+8; col = l16.
    const int rbase = (lane < 16) ? 0 : 8;
    float* zb = Z + (size_t)m0 * NOUT + nt * 16 + l16;
    #pragma unroll
    for (int r = 0; r < 8; ++r) {
      zb[(size_t)(rbase + r) * NOUT] = acc[r];
    }
  }
}

// ---------------------------------------------------------------------------
// Reduce: one wave per node. Lane e holds idx_e (E==32==wave32).
// Broadcast via readlane -> SGPR base for coalesced 512B row gathers from L2.
// ---------------------------------------------------------------------------
__device__ __forceinline__ float elu1(float z) {
  return z > 0.0f ? z : expm1f(z);
}

__global__ __launch_bounds__(256)
void edgeconv_reduce(const float* __restrict__ Z, const int* __restrict__ eidx,
                     const float* __restrict__ bias, float* __restrict__ out) {
  const int tid  = threadIdx.x;
  const int lane = tid & 31;
  const int node = blockIdx.x * 8 + (tid >> 5);
  if (node >= M_DIM) return;

  const int bbase = (node / N_DIM) * N_DIM;            // batch row base into Z

  const int myidx = eidx[(size_t)node * E_DIM + lane]; // lane e -> idx[node][e]

  const float4* zrow = (const float4*)(Z + (size_t)node * NOUT);
  const float4 z1 = zrow[lane];        // Z1 cols lane*4 .. +3
  const float4 z2 = zrow[lane + 32];   // Z2 cols 128+lane*4 ..
  const float4 bv = ((const float4*)bias)[lane];

  const float ninf = -__builtin_inff();
  float4 mx = make_float4(ninf, ninf, ninf, ninf);
  bool any = false;

  #pragma unroll 4
  for (int e = 0; e < E_DIM; ++e) {
    const int s = __builtin_amdgcn_readlane(myidx, e); // uniform (SGPR)
    if (s >= 0) {                                      // scalar branch
      any = true;
      const float4 g =
          ((const float4*)(Z + (size_t)(bbase + s) * NOUT + O_DIM))[lane];
      mx.x = fmaxf(mx.x, g.x);
      mx.y = fmaxf(mx.y, g.y);
      mx.z = fmaxf(mx.z, g.z);
      mx.w = fmaxf(mx.w, g.w);
    }
  }

  float4 r;
  if (any) {
    r.x = elu1(z1.x - z2.x + bv.x + mx.x);
    r.y = elu1(z1.y - z2.y + bv.y + mx.y);
    r.z = elu1(z1.z - z2.z + bv.z + mx.z);
    r.w = elu1(z1.w - z2.w + bv.w + mx.w);
  } else {
    r = make_float4(ninf, ninf, ninf, ninf);           // all edges masked
  }
  ((float4*)(out + (size_t)node * O_DIM))[lane] = r;
}

// ---------------------------------------------------------------------------
extern "C" void kernel_launch(void* const* d_in, const int* in_sizes, int n_in,
                              void* d_out, int out_size, void* d_ws, size_t ws_size,
                              hipStream_t stream) {
  (void)in_sizes; (void)n_in; (void)out_size; (void)ws_size;

  const float* x    = (const float*)d_in[0];   // [B,N,C] fp32
  const int*   eidx = (const int*)d_in[1];     // [B,N,E] int
  const float* W    = (const float*)d_in[2];   // [2C,O] fp32
  const float* bias = (const float*)d_in[3];   // [O] fp32
  float*       out  = (float*)d_out;           // [B,N,O] fp32

  float* Z  = (float*)d_ws;                    // [M, 256] = [Z1|Z2], ~41 MB
  v2f*   Bp = (v2f*)(Z + (size_t)M_DIM * NOUT);// packed W fragments, 128 KB

  edgeconv_packB<<<(NTILES * KSTEPS * 32 + 255) / 256, 256, 0, stream>>>(W, Bp);
  edgeconv_gemm<<<M_DIM / 16, 256, 0, stream>>>(x, Bp, Z);
  edgeconv_reduce<<<M_DIM / 8, 256, 0, stream>>>(Z, eidx, bias, out);
}